// MultiHeadLinearAttention_73761768341580
// MI455X (gfx1250) — compile-verified
//
#include <hip/hip_runtime.h>
#include <hip/hip_bf16.h>

#define HEADS 8
#define DIM_HEAD 64
#define EPS 1e-6f

typedef __attribute__((ext_vector_type(2))) float v2f;
typedef __attribute__((ext_vector_type(8))) float v8f;

// ---------------------------------------------------------------------------
// Generic fp32 WMMA GEMM:  C[M,N] = A[M,K] @ B[K,N] (+ bias[N])
// One wave per 16x64 output tile, K stepped by 4 using V_WMMA_F32_16X16X4_F32.
// Lane layouts per CDNA5 ISA 7.12.2:
//   A (16x4 f32): lane = khalf*16 + m,  a[j] = A[m][k + 2*khalf + j]
//   B (4x16 f32): lane = khalf*16 + n,  b[j] = B[k + 2*khalf + j][n]
//   C/D (16x16 f32): vgpr r: lanes0-15 -> M=r, lanes16-31 -> M=r+8; N=lane%16
// ---------------------------------------------------------------------------
__global__ __launch_bounds__(256) void wmma_gemm_f32(
    const float* __restrict__ A, const float* __restrict__ B,
    const float* __restrict__ bias, float* __restrict__ C,
    int M, int N, int K) {
  const int wave_id = (blockIdx.x * blockDim.x + threadIdx.x) >> 5; // wave32
  const int lane = threadIdx.x & 31;
  const int tilesN = N >> 6; // N / 64
  const int tm = wave_id / tilesN;
  const int tn = wave_id % tilesN;
  const int m0 = tm * 16;
  const int n0 = tn * 64;
  if (m0 >= M) return; // wave-uniform: EXEC stays all-1s for executing waves

  const int ml = lane & 15;   // A row within tile, also B column within subtile
  const int kh = lane >> 4;   // k-half (0 or 1)

  const float* __restrict__ arow = A + (size_t)(m0 + ml) * K + 2 * kh;
  const float* __restrict__ bbase = B + (size_t)(2 * kh) * N + n0 + ml;

  v8f acc0 = {}, acc1 = {}, acc2 = {}, acc3 = {};

#pragma unroll 4
  for (int k = 0; k < K; k += 4) {
    v2f a;
    a.x = arow[k];
    a.y = arow[k + 1];
    const float* bp = bbase + (size_t)k * N;

    v2f b0, b1, b2, b3;
    b0.x = bp[0];      b0.y = bp[N];
    b1.x = bp[16];     b1.y = bp[N + 16];
    b2.x = bp[32];     b2.y = bp[N + 32];
    b3.x = bp[48];     b3.y = bp[N + 48];

    acc0 = __builtin_amdgcn_wmma_f32_16x16x4_f32(false, a, false, b0, (short)0, acc0, false, false);
    acc1 = __builtin_amdgcn_wmma_f32_16x16x4_f32(false, a, false, b1, (short)0, acc1, false, false);
    acc2 = __builtin_amdgcn_wmma_f32_16x16x4_f32(false, a, false, b2, (short)0, acc2, false, false);
    acc3 = __builtin_amdgcn_wmma_f32_16x16x4_f32(false, a, false, b3, (short)0, acc3, false, false);
  }

  const int rbase = (lane < 16) ? 0 : 8;
  v8f accs[4] = {acc0, acc1, acc2, acc3};
#pragma unroll
  for (int s = 0; s < 4; ++s) {
#pragma unroll
    for (int r = 0; r < 8; ++r) {
      const int row = m0 + rbase + r;
      const int col = n0 + s * 16 + ml;
      float val = accs[s][r];
      if (bias != nullptr) val += bias[col];
      C[(size_t)row * N + col] = val;
    }
  }
}

// ---------------------------------------------------------------------------
// RoPE + feature maps.  qkv row layout: [q(512) | k(512) | v(512)], head h
// occupies cols h*64..h*64+63 within each third.  Writes phi_q, phi_k, v in
// (b, h, n, d) layout for the scan.  64 threads = one (b, t, h) row.
// ---------------------------------------------------------------------------
__global__ __launch_bounds__(64) void rope_phi_kernel(
    const float* __restrict__ qkv,
    float* __restrict__ phi_q, float* __restrict__ phi_k, float* __restrict__ vout,
    int n) {
  const int h = blockIdx.x % HEADS;
  const int t = (blockIdx.x / HEADS) % n;
  const int b = blockIdx.x / (HEADS * n);
  const int j = threadIdx.x; // 0..63

  const size_t row = (size_t)(b * n + t) * (3 * HEADS * DIM_HEAD);
  const float* qrow = qkv + row + h * DIM_HEAD;
  const float* krow = qkv + row + HEADS * DIM_HEAD + h * DIM_HEAD;
  const float* vrow = qkv + row + 2 * HEADS * DIM_HEAD + h * DIM_HEAD;

  // RoPE: inv_freq[i] = 10000^(-2i/64), emb = [freqs, freqs]
  const int jm = j & 31;
  const float inv_freq = powf(10000.0f, -(float)(2 * jm) / 64.0f);
  const float ang = (float)t * inv_freq;
  const float c = cosf(ang);
  const float s = sinf(ang);

  const float qj = qrow[j];
  const float kj = krow[j];
  const float qr = (j < 32) ? -qrow[j + 32] : qrow[j - 32];
  const float kr = (j < 32) ? -krow[j + 32] : krow[j - 32];
  const float q = qj * c + qr * s;
  const float k = kj * c + kr * s;

  // softmax over d=64 (two waves -> LDS reduction)
  __shared__ float buf[64];
  buf[j] = q;
  __syncthreads();
  float mx = -INFINITY;
#pragma unroll
  for (int i = 0; i < 64; ++i) mx = fmaxf(mx, buf[i]);
  const float e = expf(q - mx);
  __syncthreads();
  buf[j] = e;
  __syncthreads();
  float sum = 0.0f;
#pragma unroll
  for (int i = 0; i < 64; ++i) sum += buf[i];

  const float pq = (e / sum) * 0.125f; // * d^-0.5 = 1/8

  const size_t o = (((size_t)b * HEADS + h) * n + t) * DIM_HEAD + j;
  phi_q[o] = pq;
  phi_k[o] = expf(k);
  vout[o] = vrow[j];
}

// ---------------------------------------------------------------------------
// Chunked causal linear-attention scan, WMMA version.
// One single-wave block per (b, h, strip w), w in [0,4): the wave owns the
// 16-wide column strip [16w,16w+16) of the 64x64 KV state S and of the output
// d-dimension.  Per chunk of 16 timesteps:
//   scores = Q·K^T (K=64)           -> 16 wmma
//   num    = Q·S_prev (strip, K=64) -> 16 wmma
//   num   += mask(scores)·V (K=16)  ->  4 wmma   (scores via LDS relayout)
//   S strip += K^T·V (K=16)         -> 16 wmma   (C-frags staged in LDS)
//   z (replicated per wave, in regs) += colsum(K) via shfl_xor within halves
//   den = Q·z_prev + rowsum(mask(scores))
// S strip stored transposed in LDS (Sw[n][r], pitch 68 -> conflict-free).
// ---------------------------------------------------------------------------
__global__ __launch_bounds__(32) void linear_attn_scan_wmma(
    const float* __restrict__ phi_q, const float* __restrict__ phi_k,
    const float* __restrict__ v, float* __restrict__ attn, int n) {
  const int blk = blockIdx.x;   // 0 .. B*HEADS*4 - 1
  const int w = blk & 3;        // strip index
  const int bh = blk >> 2;
  const int b = bh / HEADS;
  const int h = bh % HEADS;

  const int lane = threadIdx.x & 31;
  const int hz = lane >> 4;     // k-half / row-half
  const int nl = lane & 15;     // n / m within tile
  const int rbase = hz * 8;     // C-frag row base

  __shared__ float Sw[16][68];  // Sw[n][r] = S[r][16w+n]  (padded pitch)
  __shared__ float Acm[16][17]; // masked scores, Acm[tj][ti] (padded)
  __shared__ float denI[16], denA[16];

  for (int i = lane; i < 16 * 68; i += 32) (&Sw[0][0])[i] = 0.0f;
  v2f zreg[16];
#pragma unroll
  for (int kk = 0; kk < 16; ++kk) { zreg[kk].x = 0.0f; zreg[kk].y = 0.0f; }
  __syncthreads();

  const size_t base = (size_t)bh * n * DIM_HEAD;
  const float* __restrict__ pq = phi_q + base;
  const float* __restrict__ pk = phi_k + base;
  const float* __restrict__ pv = v + base;

  for (int t0 = 0; t0 < n; t0 += 16) {
    // ---- phase 1: scores, num_inter, den_inter, z update ----
    const float* qrow = pq + (size_t)(t0 + nl) * 64 + 2 * hz;
    const float* krow = pk + (size_t)(t0 + nl) * 64 + 2 * hz;
    v2f qf[16], kf[16];
#pragma unroll
    for (int kk = 0; kk < 16; ++kk) {
      qf[kk].x = qrow[4 * kk];
      qf[kk].y = qrow[4 * kk + 1];
      kf[kk].x = krow[4 * kk];
      kf[kk].y = krow[4 * kk + 1];
    }

    // scores = Q @ K^T   (A = Q frags, B = K^T frags == kf)
    v8f sc = {};
#pragma unroll
    for (int kk = 0; kk < 16; ++kk)
      sc = __builtin_amdgcn_wmma_f32_16x16x4_f32(false, qf[kk], false, kf[kk], (short)0, sc, false, false);

    // den_inter = Q . z_prev   (per-lane partial over its 32 features)
    float p = 0.0f;
#pragma unroll
    for (int kk = 0; kk < 16; ++kk)
      p += qf[kk].x * zreg[kk].x + qf[kk].y * zreg[kk].y;
    p += __shfl_xor(p, 16, 32); // combine the two k-halves -> den_inter[row nl]

    // num_inter = Q @ S_prev (strip): B-frags contiguous from transposed LDS
    v8f acc = {};
#pragma unroll
    for (int kk = 0; kk < 16; ++kk) {
      v2f bs;
      bs.x = Sw[nl][4 * kk + 2 * hz];
      bs.y = Sw[nl][4 * kk + 2 * hz + 1];
      acc = __builtin_amdgcn_wmma_f32_16x16x4_f32(false, qf[kk], false, bs, (short)0, acc, false, false);
    }

    // z += colsum(K_chunk): reduce kf over the 16 lanes of each half
#pragma unroll
    for (int kk = 0; kk < 16; ++kk) {
#pragma unroll
      for (int st = 1; st < 16; st <<= 1) {
        kf[kk].x += __shfl_xor(kf[kk].x, st, 32);
        kf[kk].y += __shfl_xor(kf[kk].y, st, 32);
      }
      zreg[kk].x += kf[kk].x;
      zreg[kk].y += kf[kk].y;
    }

    // causal mask (keep key tj <= query ti) + den_intra row sums
    float da[8];
#pragma unroll
    for (int r = 0; r < 8; ++r) {
      if (nl > rbase + r) sc[r] = 0.0f;
      da[r] = sc[r];
    }
#pragma unroll
    for (int st = 1; st < 16; st <<= 1) {
#pragma unroll
      for (int r = 0; r < 8; ++r) da[r] += __shfl_xor(da[r], st, 32);
    }

    // stage masked scores (col-major by tj) + dens into LDS
#pragma unroll
    for (int r = 0; r < 8; ++r) Acm[nl][rbase + r] = sc[r];
    if (lane == 0) {
#pragma unroll
      for (int r = 0; r < 8; ++r) denA[r] = da[r];
    }
    if (lane == 16) {
#pragma unroll
      for (int r = 0; r < 8; ++r) denA[8 + r] = da[r];
    }
    if (lane < 16) denI[nl] = p;
    __syncthreads(); // single-wave: acts as LDS wait + code-motion barrier

    // ---- phase 2: num_intra, S update, output ----
    v2f vf[4];
#pragma unroll
    for (int s = 0; s < 4; ++s) {
      const int kidx = 4 * s + 2 * hz;
      vf[s].x = pv[(size_t)(t0 + kidx) * 64 + 16 * w + nl];
      vf[s].y = pv[(size_t)(t0 + kidx + 1) * 64 + 16 * w + nl];
      v2f af; // A-frag of masked scores: Am[m][kidx+j] = Acm[kidx+j][m]
      af.x = Acm[kidx][nl];
      af.y = Acm[kidx + 1][nl];
      acc = __builtin_amdgcn_wmma_f32_16x16x4_f32(false, af, false, vf[s], (short)0, acc, false, false);
    }

    // S strip += K^T @ V  (4 row-tiles, C-frags staged through transposed LDS)
#pragma unroll
    for (int kt = 0; kt < 4; ++kt) {
      v8f cS;
#pragma unroll
      for (int r = 0; r < 8; ++r) cS[r] = Sw[nl][16 * kt + rbase + r];
#pragma unroll
      for (int s = 0; s < 4; ++s) {
        const int kidx = 4 * s + 2 * hz;
        v2f ka; // A-frag of K^T: KT[16kt+m][kidx+j] = K[kidx+j][16kt+m]
        ka.x = pk[(size_t)(t0 + kidx) * 64 + 16 * kt + nl];
        ka.y = pk[(size_t)(t0 + kidx + 1) * 64 + 16 * kt + nl];
        cS = __builtin_amdgcn_wmma_f32_16x16x4_f32(false, ka, false, vf[s], (short)0, cS, false, false);
      }
#pragma unroll
      for (int r = 0; r < 8; ++r) Sw[nl][16 * kt + rbase + r] = cS[r];
    }

    // output:  attn[(b, t0+ti, h*64 + 16w + nl)] = num / max(den, EPS)
#pragma unroll
    for (int r = 0; r < 8; ++r) {
      const int ti = rbase + r;
      const float den = fmaxf(denI[ti] + denA[ti], EPS);
      attn[(size_t)(b * n + t0 + ti) * (HEADS * DIM_HEAD) + h * DIM_HEAD + 16 * w + nl] =
          acc[r] / den;
    }
    __syncthreads();
  }
}

// ---------------------------------------------------------------------------
extern "C" void kernel_launch(void* const* d_in, const int* in_sizes, int n_in,
                              void* d_out, int out_size, void* d_ws, size_t ws_size,
                              hipStream_t stream) {
  (void)in_sizes; (void)n_in; (void)out_size; (void)ws_size;

  const float* x     = (const float*)d_in[0]; // (2, 2048, 512)
  const float* W_qkv = (const float*)d_in[1]; // (512, 1536)
  const float* W_out = (const float*)d_in[2]; // (512, 512)
  const float* b_out = (const float*)d_in[3]; // (512,)
  float* out = (float*)d_out;                 // (2, 2048, 512)

  const int B = 2, N = 2048, DIM = 512;
  const int INNER = HEADS * DIM_HEAD; // 512
  const int M = B * N;                // 4096 rows
  const int QKV_N = 3 * INNER;        // 1536

  // Workspace carve-up (bytes):
  //   [0, 25165824)            qkv   (4096*1536 f32)  -- later reused for attn
  //   [25165824, +8388608)     phi_q (2*8*2048*64 f32)
  //   next                     phi_k
  //   next                     v
  char* ws = (char*)d_ws;
  float* qkv_ws  = (float*)(ws);
  float* attn_ws = (float*)(ws); // reuses qkv region (qkv dead after rope_phi)
  size_t off = (size_t)M * QKV_N * sizeof(float);
  float* phiq_ws = (float*)(ws + off); off += (size_t)B * HEADS * N * DIM_HEAD * sizeof(float);
  float* phik_ws = (float*)(ws + off); off += (size_t)B * HEADS * N * DIM_HEAD * sizeof(float);
  float* v_ws    = (float*)(ws + off);

  // 1) qkv = x @ W_qkv   (M=4096, N=1536, K=512)
  {
    const int waves = (M / 16) * (QKV_N / 64); // 6144
    const int blocks = (waves * 32 + 255) / 256;
    wmma_gemm_f32<<<blocks, 256, 0, stream>>>(x, W_qkv, nullptr, qkv_ws, M, QKV_N, DIM);
  }

  // 2) RoPE + phi maps, re-layout to (b, h, n, d)
  rope_phi_kernel<<<B * N * HEADS, 64, 0, stream>>>(qkv_ws, phiq_ws, phik_ws, v_ws, N);

  // 3) chunked causal linear-attention scan: one single-wave block per
  //    (b, h, 16-wide d-strip) -> 64 independent blocks, all on WMMA
  linear_attn_scan_wmma<<<B * HEADS * 4, 32, 0, stream>>>(phiq_ws, phik_ws, v_ws, attn_ws, N);

  // 4) out = attn @ W_out + b_out   (M=4096, N=512, K=512)
  {
    const int waves = (M / 16) * (INNER / 64); // 2048
    const int blocks = (waves * 32 + 255) / 256;
    wmma_gemm_f32<<<blocks, 256, 0, stream>>>(attn_ws, W_out, b_out, out, M, INNER, DIM);
  }
}